// ClusterModel_82033875354199
// MI455X (gfx1250) — compile-verified
//
#include <hip/hip_runtime.h>
#include <math.h>

typedef __attribute__((ext_vector_type(2))) float v2f;
typedef __attribute__((ext_vector_type(8))) float v8f;

#define B_    16
#define D_    384
#define H_    64
#define W_    64
#define HW    (H_ * W_)        // 4096
#define NPIX  (B_ * HW)        // 65536
#define NC    8                // pca components
#define NK    14               // clusters
#define KCH   (D_ / 4)         // 96 K-chunks of 4

// workspace layout (in floats)
#define WS_Y     0                       // 65536 * 8 projected vectors
#define WS_COMP  (NPIX * NC)             // 96*32 v2f = 6144 floats (A-tile pack)
#define WS_MU    (WS_COMP + KCH * 64)    // 8 floats: mean @ components^T
#define WS_CN    (WS_MU + NC)            // 14 floats: ||centroid_k||^2

// ---------------------------------------------------------------------------
// Setup: pack pca_components into the WMMA A-operand lane layout (16x4 f32
// tile, rows 8..15 zero-padded), precompute projected mean and centroid norms.
// A layout (32-bit A 16x4): lanes 0-15 hold {K=k0,K=k0+1} for M=lane,
// lanes 16-31 hold {K=k0+2,K=k0+3} for M=lane-16.
// ---------------------------------------------------------------------------
__global__ void __launch_bounds__(256) setup_k(const float* __restrict__ mean,
                                               const float* __restrict__ comp,
                                               const float* __restrict__ cent,
                                               float* __restrict__ ws) {
  const int t = threadIdx.x;
  for (int i = t; i < KCH * 32; i += 256) {
    const int c  = i >> 5;
    const int l  = i & 31;
    const int m  = l & 15;
    const int kk = c * 4 + ((l >> 4) << 1);
    float v0 = 0.0f, v1 = 0.0f;
    if (m < NC) { v0 = comp[m * D_ + kk]; v1 = comp[m * D_ + kk + 1]; }
    ws[WS_COMP + i * 2 + 0] = v0;
    ws[WS_COMP + i * 2 + 1] = v1;
  }
  if (t < NC) {
    float s = 0.0f;
    for (int d = 0; d < D_; ++d) s += mean[d] * comp[t * D_ + d];
    ws[WS_MU + t] = s;
  }
  if (t >= 32 && t < 32 + NK) {
    const int k = t - 32;
    float s = 0.0f;
    for (int j = 0; j < NC; ++j) { const float v = cent[k * NC + j]; s += v * v; }
    ws[WS_CN + k] = s;
  }
}

// ---------------------------------------------------------------------------
// Projection GEMM on the ORIGINAL 64x64 grid (before upsampling):
// Y[pixel, c] = sum_d ssl[b, d, h, w] * comp[c, d], via V_WMMA_F32_16X16X4_F32.
// A = padded components tile (M=16 comps x K=4), B = embeddings (K=4 x N=16 px).
// B-operand lane layout falls out of channels-first memory for free:
//   VGPR0: lanes0-15 K=k0, lanes16-31 K=k0+2;  VGPR1: K=k0+1 / K=k0+3.
// Each wave owns 32 pixels (two N-tiles), 96 K-chunks, 192 WMMAs.
// D layout: lanes 0-15, VGPR j = D[M=j][N=lane] -> lane holds all 8
// components of its pixel contiguously -> one v8f store.
// ---------------------------------------------------------------------------
__global__ void __launch_bounds__(256) proj_k(const float* __restrict__ ssl,
                                              float* __restrict__ ws) {
  const int lane  = threadIdx.x & 31;
  const int wave  = blockIdx.x * (blockDim.x >> 5) + (threadIdx.x >> 5);
  const int pbase = wave * 32;                       // 32 pixels per wave
  const int batch = pbase >> 12;                     // / 4096
  const int hw    = pbase & (HW - 1);
  const int col   = lane & 15;
  const int krow  = (lane >> 4) << 1;                // 0 or 2

  const float* sb = ssl + (size_t)batch * D_ * HW + (size_t)krow * HW + hw + col;
  const v2f*   aw = (const v2f*)(ws + WS_COMP);

  v8f acc0 = {};
  v8f acc1 = {};

#pragma unroll 4
  for (int c = 0; c < KCH; ++c) {
    const v2f a = aw[c * 32 + lane];                 // coalesced b64
    const float* p = sb + (size_t)c * 4 * HW;
    v2f b0, b1;
    b0.x = p[0];        b0.y = p[HW];                // N-tile 0
    b1.x = p[16];       b1.y = p[16 + HW];           // N-tile 1
    acc0 = __builtin_amdgcn_wmma_f32_16x16x4_f32(false, a, false, b0,
                                                 (short)0, acc0, false, false);
    acc1 = __builtin_amdgcn_wmma_f32_16x16x4_f32(false, a, false, b1,
                                                 (short)0, acc1, false, false);
  }

  if (lane < 16) {
    *(v8f*)(ws + WS_Y + (size_t)(pbase + lane) * NC)      = acc0;
    *(v8f*)(ws + WS_Y + (size_t)(pbase + 16 + lane) * NC) = acc1;
  }
}

// ---------------------------------------------------------------------------
// Upsample (half-pixel bilinear, 2x) the 8-dim projected vectors, subtract
// projected mean, normalize (folded into the dot products), argmin over 14
// centroids. First-min tie semantics match jnp.argmin.
// ---------------------------------------------------------------------------
__global__ void __launch_bounds__(256) label_k(const float* __restrict__ ws,
                                               const float* __restrict__ cent,
                                               float* __restrict__ out) {
  __shared__ __align__(32) float sc[NK * NC];
  __shared__ __align__(32) float smu[NC];
  __shared__ float scn[NK];
  const int t = threadIdx.x;
  if (t < NK * NC)              sc[t]        = cent[t];
  if (t >= 128 && t < 128 + NC) smu[t - 128] = ws[WS_MU + (t - 128)];
  if (t >= 160 && t < 160 + NK) scn[t - 160] = ws[WS_CN + (t - 160)];
  __syncthreads();

  const int idx = blockIdx.x * 256 + t;              // 262144 outputs
  const int b   = idx >> 14;
  const int rem = idx & 16383;
  const int y   = rem >> 7;
  const int x   = rem & 127;

  const float sy = 0.5f * (float)y - 0.25f;
  const float sx = 0.5f * (float)x - 0.25f;
  const float fy = floorf(sy), fx = floorf(sx);
  const float wy = sy - fy,    wx = sx - fx;
  const int y0 = max(0, min(H_ - 1, (int)fy));
  const int y1 = max(0, min(H_ - 1, (int)fy + 1));
  const int x0 = max(0, min(W_ - 1, (int)fx));
  const int x1 = max(0, min(W_ - 1, (int)fx + 1));

  const v8f* Y = (const v8f*)(ws + WS_Y);
  const int base = b * HW;
  const v8f c00 = Y[base + y0 * W_ + x0];
  const v8f c01 = Y[base + y0 * W_ + x1];
  const v8f c10 = Y[base + y1 * W_ + x0];
  const v8f c11 = Y[base + y1 * W_ + x1];

  v8f r = c00 * ((1.0f - wy) * (1.0f - wx))
        + c01 * ((1.0f - wy) * wx)
        + c10 * (wy * (1.0f - wx))
        + c11 * (wy * wx);
  r -= *(const v8f*)smu;

  float n2 = 0.0f;
#pragma unroll
  for (int j = 0; j < NC; ++j) n2 += r[j] * r[j];
  const float inv = 1.0f / fmaxf(sqrtf(n2), 1e-12f);

  int best = 0;
  float bd = 3.402823e38f;
#pragma unroll
  for (int k = 0; k < NK; ++k) {
    float dot = 0.0f;
#pragma unroll
    for (int j = 0; j < NC; ++j) dot += r[j] * sc[k * NC + j];
    const float d = scn[k] - 2.0f * inv * dot;       // ||r̂||² constant per pixel
    if (d < bd) { bd = d; best = k; }
  }
  out[idx] = (float)best;
}

// ---------------------------------------------------------------------------
extern "C" void kernel_launch(void* const* d_in, const int* in_sizes, int n_in,
                              void* d_out, int out_size, void* d_ws, size_t ws_size,
                              hipStream_t stream) {
  const float* ssl  = (const float*)d_in[0];   // [16,384,64,64]
  const float* mean = (const float*)d_in[1];   // [384]
  const float* comp = (const float*)d_in[2];   // [8,384]
  const float* cent = (const float*)d_in[3];   // [14,8]
  float* ws  = (float*)d_ws;
  float* out = (float*)d_out;                  // [16,1,128,128]

  setup_k<<<1, 256, 0, stream>>>(mean, comp, cent, ws);
  proj_k<<<(NPIX / 32) / 8, 256, 0, stream>>>(ssl, ws);      // 256 blocks, 8 waves each
  label_k<<<(B_ * 4 * HW) / 256, 256, 0, stream>>>(ws, cent, out);
}